// BiMambaBlock_70574902608474
// MI455X (gfx1250) — compile-verified
//
#include <hip/hip_runtime.h>
#include <math.h>

// Problem constants (from reference)
#define D_MODELC 512
#define D_INNERC 1024
#define D_STATEC 16
#define DT_RANKC 32
#define NB 2
#define SEQ 2048
#define BLROWS (NB * SEQ)        // 4096
#define CHUNK_LEN 128
#define NCHUNK (SEQ / CHUNK_LEN) // 16

// GEMM tiling
#define BM 128                   // output rows per 256-thread block (8 waves x 16)
#define BN 64                    // output cols per block (4 wmma n-tiles per wave)
#define KT 32                    // K-stage per LDS tile
#define KPAIRS (KT / 2)          // 16 interleaved K-pair rows
#define BNPAIR 80                // padded pair-row stride (pairs): 640B per row ->
                                 // +32 banks/row, lanes 16-31 disjoint from 0-15

typedef __attribute__((ext_vector_type(2))) float v2f;
typedef __attribute__((ext_vector_type(8))) float v8f;

__device__ __forceinline__ float siluf(float x) { return x / (1.f + __expf(-x)); }

// ---------------------------------------------------------------------------
// Blocked f32 GEMM on V_WMMA_F32_16X16X4_F32, double-buffered LDS weight tile
// stored K-pair-interleaved: element (k,n) -> lb[k/2][n].{x=even k, y=odd k}.
// The wmma B fragment (rows {k+2*half, k+2*half+1} at col n) is then a single
// aligned ds_load_b64 with no register repacking.
// C[M,N] = act(A[M,K] @ W[K,N] + bias).
// Block: 256 thr = 8 waves -> 128x64 tile; wave -> 16x64 (4 accumulators).
// flipL: A rows read per-batch sequence-reversed (fuses jnp.flip).
// Fragment layouts per ISA 7.12.2:
//   A 16x4: lanes 0-15 K={0,1} in vgpr{0,1}; lanes 16-31 K={2,3}.
//   B 4x16: vgpr0 = rows {0,2}, vgpr1 = rows {1,3}, col = lane&15.
//   C/D:    vgpr v -> row m0 + v + 8*(lane>=16), col = lane&15.
// ---------------------------------------------------------------------------
__global__ __launch_bounds__(256) void wmma_gemm_f32(
    const float* __restrict__ A, int lda,
    const float* __restrict__ W, int ldw,
    float* __restrict__ C, int ldc,
    int M, int N, int K,
    int flipL, int Lrow,
    const float* __restrict__ bias, int act)
{
    __shared__ v2f lb[2][KPAIRS][BNPAIR];         // 2 x 10KB

    const int tid  = threadIdx.x;
    const int lane = tid & 31;
    const int wv   = tid >> 5;                    // wave in block: 0..7
    const int nb   = N / BN;
    const int bm0  = (blockIdx.x / nb) * BM;
    const int n00  = (blockIdx.x % nb) * BN;
    const int half = lane >> 4;                   // 0: lanes 0-15, 1: 16-31
    const int ln   = lane & 15;

    // Per-lane A row (optionally sequence-flipped within its batch)
    int arow = bm0 + wv * 16 + ln;
    if (flipL) {
        int bb = arow / Lrow;
        int ll = arow - bb * Lrow;
        arow = bb * Lrow + (Lrow - 1 - ll);
    }
    const float* __restrict__ Ar = A + (size_t)arow * lda + 2 * half;

    // Cooperative staging: thread -> K-pair row j (2 global rows) x 4 columns.
    const int j  = tid >> 4;                      // pair row 0..15
    const int cg = (tid & 15) * 4;                // column group
    const float* __restrict__ Wg = W + (size_t)(2 * j) * ldw + n00 + cg;

    v8f acc[4];
    #pragma unroll
    for (int s = 0; s < 4; ++s)
        #pragma unroll
        for (int v = 0; v < 8; ++v) acc[s][v] = 0.f;

    // Preload stage 0: rows 2j, 2j+1; interleave into pair layout
    #pragma unroll
    for (int i = 0; i < 4; ++i) {
        v2f p;
        p.x = Wg[i];                              // W[2j  ][n00+cg+i]
        p.y = Wg[(size_t)ldw + i];                // W[2j+1][n00+cg+i]
        lb[0][j][cg + i] = p;
    }
    __syncthreads();

    int buf = 0;
    for (int k0 = 0; k0 < K; k0 += KT) {
        // Stage next W tile into the other buffer while computing this one
        if (k0 + KT < K) {
            const float* __restrict__ Wn = Wg + (size_t)(k0 + KT) * ldw;
            __builtin_prefetch(Ar + k0 + KT, 0, 3);      // global_prefetch_b8
            #pragma unroll
            for (int i = 0; i < 4; ++i) {
                v2f p;
                p.x = Wn[i];
                p.y = Wn[(size_t)ldw + i];
                lb[buf ^ 1][j][cg + i] = p;
            }
            if (k0 + 2 * KT < K)
                __builtin_prefetch(Wg + (size_t)(k0 + 2 * KT) * ldw, 0, 3);
        }
        #pragma unroll
        for (int kk = 0; kk < KT; kk += 4) {
            v2f a;
            a.x = Ar[k0 + kk];                    // A[m][k + 2*half]
            a.y = Ar[k0 + kk + 1];                // A[m][k + 2*half + 1]
            const int jj = (kk >> 1) + half;      // pair row for this half
            #pragma unroll
            for (int s = 0; s < 4; ++s) {
                v2f b = lb[buf][jj][s * 16 + ln]; // one ds_load_b64
                acc[s] = __builtin_amdgcn_wmma_f32_16x16x4_f32(
                    /*neg_a=*/false, a, /*neg_b=*/false, b,
                    /*c_mod=*/(short)0, acc[s],
                    /*reuse_a=*/false, /*reuse_b=*/false);
            }
        }
        __syncthreads();
        buf ^= 1;
    }

    // Epilogue: bias + optional softplus, write 16x64 per wave
    const int crow0 = bm0 + wv * 16 + 8 * half;
    #pragma unroll
    for (int s = 0; s < 4; ++s) {
        const int   ccol = n00 + s * 16 + ln;
        const float bv   = bias ? bias[ccol] : 0.f;
        #pragma unroll
        for (int v = 0; v < 8; ++v) {
            float val = acc[s][v] + bv;
            if (act == 1)  // softplus (dt projection epilogue)
                val = (val > 20.f) ? val : log1pf(__expf(val));
            C[(size_t)(crow0 + v) * ldc + ccol] = val;
        }
    }
}

// ---------------------------------------------------------------------------
// Depthwise causal conv (width 4) over L + SiLU.  xi = xz[..., :1024].
// One thread per (b,l,d); consecutive threads -> consecutive d (coalesced).
// ---------------------------------------------------------------------------
__global__ void conv_silu_kernel(const float* __restrict__ xz,
                                 const float* __restrict__ cw,
                                 const float* __restrict__ cb,
                                 float* __restrict__ xc)
{
    const int idx = blockIdx.x * blockDim.x + threadIdx.x;   // BLROWS * D_INNER
    const int d = idx & (D_INNERC - 1);
    const int r = idx >> 10;
    const int l = r & (SEQ - 1);
    float acc = cb[d];
    #pragma unroll
    for (int k = 0; k < 4; ++k) {
        int ls = l - 3 + k;
        if (ls >= 0)
            acc += cw[d * 4 + k] * xz[(size_t)(r - 3 + k) * (2 * D_INNERC) + d];
    }
    xc[(size_t)r * D_INNERC + d] = siluf(acc);
}

// ---------------------------------------------------------------------------
// Chunked selective scan, phase 1: per (b, d, chunk) compute the chunk's
// cumulative decay prod(dA) and its h-end (starting from h=0), 16 states.
// ---------------------------------------------------------------------------
__global__ void scan_chunk_kernel(const float* __restrict__ dt,
                                  const float* __restrict__ xc,
                                  const float* __restrict__ dbc,
                                  const float* __restrict__ A_log,
                                  float* __restrict__ cumA,
                                  float* __restrict__ hend)
{
    const int tid = blockIdx.x * blockDim.x + threadIdx.x;   // NB*NCHUNK*D_INNER
    const int d = tid & (D_INNERC - 1);
    const int t2 = tid >> 10;
    const int chunk = t2 & (NCHUNK - 1);
    const int b = t2 >> 4;                                    // log2(NCHUNK)=4

    float negA[D_STATEC], h[D_STATEC], cA[D_STATEC];
    #pragma unroll
    for (int n = 0; n < D_STATEC; ++n) {
        negA[n] = -__expf(A_log[d * D_STATEC + n]);
        h[n] = 0.f; cA[n] = 1.f;
    }
    const int r0 = b * SEQ + chunk * CHUNK_LEN;
    for (int t = 0; t < CHUNK_LEN; ++t) {
        const int r = r0 + t;
        const float dtv = dt[(size_t)r * D_INNERC + d];
        const float dx  = dtv * xc[(size_t)r * D_INNERC + d];
        const float* __restrict__ Bc = dbc + (size_t)r * 64 + DT_RANKC;
        #pragma unroll
        for (int n = 0; n < D_STATEC; ++n) {
            float dA = __expf(dtv * negA[n]);
            h[n]  = dA * h[n] + dx * Bc[n];
            cA[n] *= dA;
        }
    }
    const size_t base = (size_t)(b * NCHUNK + chunk) * D_STATEC * D_INNERC + d;
    #pragma unroll
    for (int n = 0; n < D_STATEC; ++n) {
        cumA[base + (size_t)n * D_INNERC] = cA[n];
        hend[base + (size_t)n * D_INNERC] = h[n];
    }
}

// ---------------------------------------------------------------------------
// Phase 2: per (b, d) sequentially combine NCHUNK chunk summaries into the
// initial state hinit for every chunk (exclusive prefix).
// ---------------------------------------------------------------------------
__global__ void scan_prefix_kernel(const float* __restrict__ cumA,
                                   const float* __restrict__ hend,
                                   float* __restrict__ hinit)
{
    const int tid = blockIdx.x * blockDim.x + threadIdx.x;   // NB * D_INNER
    const int d = tid & (D_INNERC - 1);
    const int b = tid >> 10;
    float h[D_STATEC];
    #pragma unroll
    for (int n = 0; n < D_STATEC; ++n) h[n] = 0.f;
    for (int c = 0; c < NCHUNK; ++c) {
        const size_t base = (size_t)(b * NCHUNK + c) * D_STATEC * D_INNERC + d;
        #pragma unroll
        for (int n = 0; n < D_STATEC; ++n) {
            const size_t ix = base + (size_t)n * D_INNERC;
            hinit[ix] = h[n];
            h[n] = cumA[ix] * h[n] + hend[ix];
        }
    }
}

// ---------------------------------------------------------------------------
// Phase 3: replay each chunk from its correct initial state; fuse the C
// contraction, D-skip and SiLU(z) gate so h is never materialized.
// ---------------------------------------------------------------------------
__global__ void scan_out_kernel(const float* __restrict__ dt,
                                const float* __restrict__ xc,
                                const float* __restrict__ dbc,
                                const float* __restrict__ xz,
                                const float* __restrict__ A_log,
                                const float* __restrict__ Dskip,
                                const float* __restrict__ hinit,
                                float* __restrict__ y)
{
    const int tid = blockIdx.x * blockDim.x + threadIdx.x;
    const int d = tid & (D_INNERC - 1);
    const int t2 = tid >> 10;
    const int chunk = t2 & (NCHUNK - 1);
    const int b = t2 >> 4;

    float negA[D_STATEC], h[D_STATEC];
    const size_t base = (size_t)(b * NCHUNK + chunk) * D_STATEC * D_INNERC + d;
    #pragma unroll
    for (int n = 0; n < D_STATEC; ++n) {
        negA[n] = -__expf(A_log[d * D_STATEC + n]);
        h[n] = hinit[base + (size_t)n * D_INNERC];
    }
    const float dsk = Dskip[d];
    const int r0 = b * SEQ + chunk * CHUNK_LEN;
    for (int t = 0; t < CHUNK_LEN; ++t) {
        const int r = r0 + t;
        const float dtv = dt[(size_t)r * D_INNERC + d];
        const float xcv = xc[(size_t)r * D_INNERC + d];
        const float dx  = dtv * xcv;
        const float* __restrict__ Bc = dbc + (size_t)r * 64 + DT_RANKC;
        const float* __restrict__ Cc = Bc + D_STATEC;
        float yv = 0.f;
        #pragma unroll
        for (int n = 0; n < D_STATEC; ++n) {
            float dA = __expf(dtv * negA[n]);
            h[n] = dA * h[n] + dx * Bc[n];
            yv += h[n] * Cc[n];
        }
        yv += xcv * dsk;
        const float zv = xz[(size_t)r * (2 * D_INNERC) + D_INNERC + d];
        y[(size_t)r * D_INNERC + d] = yv * siluf(zv);
    }
}

// ---------------------------------------------------------------------------
extern "C" void kernel_launch(void* const* d_in, const int* in_sizes, int n_in,
                              void* d_out, int out_size, void* d_ws, size_t ws_size,
                              hipStream_t stream)
{
    (void)in_sizes; (void)n_in; (void)out_size; (void)ws_size;

    const float* hidden = (const float*)d_in[0];
    const float* fuse_w = (const float*)d_in[19];
    const float* fuse_b = (const float*)d_in[20];

    // Workspace layout (floats) — per-direction buffers reused across dirs.
    const size_t XZ  = (size_t)BLROWS * 2 * D_INNERC;                // 8.39M
    const size_t XC  = (size_t)BLROWS * D_INNERC;                    // 4.19M
    const size_t DBC = (size_t)BLROWS * 64;                          // 0.26M
    const size_t CS  = (size_t)NB * NCHUNK * D_STATEC * D_INNERC;    // 0.52M
    float* xzb  = (float*)d_ws;
    float* xcb  = xzb  + XZ;
    float* dbcb = xcb  + XC;
    float* dtb  = dbcb + DBC;
    float* yb   = dtb  + XC;
    float* cA   = yb   + XC;
    float* hE   = cA   + CS;
    float* hI   = hE   + CS;
    float* ocat = hI   + CS;   // (BLROWS, 1024): cols 0-511 fwd, 512-1023 flipped-bwd

    auto gemm = [&](const float* A, int lda, const float* W, int ldw,
                    float* C, int ldc, int M, int N, int K,
                    int flip, const float* bias, int act) {
        int blocks = (M / BM) * (N / BN);         // dims divide exactly
        wmma_gemm_f32<<<blocks, 256, 0, stream>>>(A, lda, W, ldw, C, ldc,
                                                  M, N, K, flip, SEQ, bias, act);
    };

    for (int dir = 0; dir < 2; ++dir) {
        const int p = 1 + dir * 9;
        const float* in_w    = (const float*)d_in[p + 0];
        const float* conv_w  = (const float*)d_in[p + 1];
        const float* conv_b  = (const float*)d_in[p + 2];
        const float* xproj_w = (const float*)d_in[p + 3];
        const float* dt_w    = (const float*)d_in[p + 4];
        const float* dt_bias = (const float*)d_in[p + 5];
        const float* A_log   = (const float*)d_in[p + 6];
        const float* D_skip  = (const float*)d_in[p + 7];
        const float* out_w   = (const float*)d_in[p + 8];

        // 1) in-projection (bwd reads x sequence-flipped): xz = flip?(x) @ in_w
        gemm(hidden, D_MODELC, in_w, 2 * D_INNERC, xzb, 2 * D_INNERC,
             BLROWS, 2 * D_INNERC, D_MODELC, dir, nullptr, 0);
        // 2) depthwise causal conv + SiLU
        conv_silu_kernel<<<(BLROWS * D_INNERC) / 256, 256, 0, stream>>>(
            xzb, conv_w, conv_b, xcb);
        // 3) x-projection: dbc = xc @ xproj_w   (dt_raw | B | C)
        gemm(xcb, D_INNERC, xproj_w, 64, dbcb, 64,
             BLROWS, 64, D_INNERC, 0, nullptr, 0);
        // 4) dt-projection + bias + softplus: dt = softplus(dbc[:, :32] @ dt_w + b)
        gemm(dbcb, 64, dt_w, D_INNERC, dtb, D_INNERC,
             BLROWS, D_INNERC, DT_RANKC, 0, dt_bias, 1);
        // 5-7) chunked selective scan + fused y epilogue
        scan_chunk_kernel<<<(NB * NCHUNK * D_INNERC) / 256, 256, 0, stream>>>(
            dtb, xcb, dbcb, A_log, cA, hE);
        scan_prefix_kernel<<<(NB * D_INNERC) / 256, 256, 0, stream>>>(cA, hE, hI);
        scan_out_kernel<<<(NB * NCHUNK * D_INNERC) / 256, 256, 0, stream>>>(
            dtb, xcb, dbcb, xzb, A_log, D_skip, hI, yb);
        // 8) out-projection into concat buffer; bwd reads y flipped (= flip(out))
        gemm(yb, D_INNERC, out_w, D_MODELC, ocat + dir * D_MODELC, 2 * D_MODELC,
             BLROWS, D_MODELC, D_INNERC, dir, nullptr, 0);
    }

    // 9) fuse: out = concat(out_fwd, out_rev) @ fuse_w + fuse_b
    gemm(ocat, 2 * D_MODELC, fuse_w, D_MODELC, (float*)d_out, D_MODELC,
         BLROWS, D_MODELC, 2 * D_MODELC, 0, fuse_b, 0);
}